// GaussianSplattingRenderer_52750788329717
// MI455X (gfx1250) — compile-verified
//
#include <hip/hip_runtime.h>

typedef __attribute__((ext_vector_type(2))) float v2f;
typedef __attribute__((ext_vector_type(8))) float v8f;

__device__ __forceinline__ float sig_(float x) { return 1.0f / (1.0f + __expf(-x)); }

// ---------------------------------------------------------------------------
// Phase 1: per-gaussian projection -> rank-6 power coefficients + color/alpha
// power(p,n) = f_p . g_n,  f_p = [X^2, Y^2, XY, X, Y, 1] (X,Y centered at cx,cy)
// ---------------------------------------------------------------------------
__global__ void gs_prep(const float* __restrict__ mean,
                        const float* __restrict__ qvec,
                        const float* __restrict__ svec_raw,
                        const float* __restrict__ color_raw,
                        const float* __restrict__ alpha_raw,
                        const float* __restrict__ viewmat,
                        const float* __restrict__ intrins,
                        float* __restrict__ wz,
                        float* __restrict__ wgc,   // [n][8] coeffs
                        float* __restrict__ wca,   // [n][4] rgb + alpha
                        int n)
{
    int i = blockIdx.x * blockDim.x + threadIdx.x;
    if (i >= n) return;
    const float fx = intrins[0], fy = intrins[1];
    float R00 = viewmat[0], R01 = viewmat[1], R02 = viewmat[2],  t0 = viewmat[3];
    float R10 = viewmat[4], R11 = viewmat[5], R12 = viewmat[6],  t1 = viewmat[7];
    float R20 = viewmat[8], R21 = viewmat[9], R22 = viewmat[10], t2 = viewmat[11];
    float mx = mean[3*i+0], my = mean[3*i+1], mz = mean[3*i+2];
    float x = R00*mx + R01*my + R02*mz + t0;
    float y = R10*mx + R11*my + R12*mz + t1;
    float z = R20*mx + R21*my + R22*mz + t2;
    float zc = fmaxf(z, 0.001f);

    float qw = qvec[4*i+0], qx = qvec[4*i+1], qy = qvec[4*i+2], qz = qvec[4*i+3];
    float qn = rsqrtf(qw*qw + qx*qx + qy*qy + qz*qz);
    qw *= qn; qx *= qn; qy *= qn; qz *= qn;
    float Q00 = 1.f - 2.f*(qy*qy + qz*qz), Q01 = 2.f*(qx*qy - qw*qz), Q02 = 2.f*(qx*qz + qw*qy);
    float Q10 = 2.f*(qx*qy + qw*qz), Q11 = 1.f - 2.f*(qx*qx + qz*qz), Q12 = 2.f*(qy*qz - qw*qx);
    float Q20 = 2.f*(qx*qz - qw*qy), Q21 = 2.f*(qy*qz + qw*qx), Q22 = 1.f - 2.f*(qx*qx + qy*qy);
    float s0 = __expf(svec_raw[3*i+0]), s1 = __expf(svec_raw[3*i+1]), s2 = __expf(svec_raw[3*i+2]);

    float j00 = fx / zc, j02 = -fx * x / (zc*zc);
    float j11 = fy / zc, j12 = -fy * y / (zc*zc);
    // T2 = J @ Rw  (2x3)
    float T00 = j00*R00 + j02*R20, T01 = j00*R01 + j02*R21, T02 = j00*R02 + j02*R22;
    float T10 = j11*R10 + j12*R20, T11 = j11*R11 + j12*R21, T12 = j11*R12 + j12*R22;
    // TM = T2 @ (Rq * diag(s))  (2x3)
    float TM00 = (T00*Q00 + T01*Q10 + T02*Q20) * s0;
    float TM01 = (T00*Q01 + T01*Q11 + T02*Q21) * s1;
    float TM02 = (T00*Q02 + T01*Q12 + T02*Q22) * s2;
    float TM10 = (T10*Q00 + T11*Q10 + T12*Q20) * s0;
    float TM11 = (T10*Q01 + T11*Q11 + T12*Q21) * s1;
    float TM12 = (T10*Q02 + T11*Q12 + T12*Q22) * s2;
    float covA = TM00*TM00 + TM01*TM01 + TM02*TM02 + 0.3f;
    float covB = TM00*TM10 + TM01*TM11 + TM02*TM12;
    float covC = TM10*TM10 + TM11*TM11 + TM12*TM12 + 0.3f;
    float det  = covA*covC - covB*covB;
    float inv  = 1.0f / fmaxf(det, 1e-8f);
    float conA = covC * inv, conB = -covB * inv, conC = covA * inv;
    float U = fx * x / zc;   // = u - cx
    float V = fy * y / zc;   // = v - cy
    bool valid = (z > 0.2f) && (det > 0.0f);
    float alpha = valid ? sig_(alpha_raw[i]) : 0.0f;   // invalid -> zero contribution

    wz[i] = z;
    float* g = wgc + i*8;
    g[0] = -0.5f * conA;
    g[1] = -0.5f * conC;
    g[2] = -conB;
    g[3] = conA*U + conB*V;
    g[4] = conC*V + conB*U;
    g[5] = -0.5f*conA*U*U - 0.5f*conC*V*V - conB*U*V;
    g[6] = 0.0f; g[7] = 0.0f;
    float* c4 = wca + i*4;
    c4[0] = sig_(color_raw[3*i+0]);
    c4[1] = sig_(color_raw[3*i+1]);
    c4[2] = sig_(color_raw[3*i+2]);
    c4[3] = alpha;
}

// ---------------------------------------------------------------------------
// Phase 2: stable depth sort via O(N^2) rank computation (N=2048, L2-resident)
// ---------------------------------------------------------------------------
__global__ void gs_sort(const float* __restrict__ wz,
                        const float* __restrict__ gc_in,
                        const float* __restrict__ ca_in,
                        float* __restrict__ gc_out,
                        float* __restrict__ ca_out,
                        int n)
{
    int i = blockIdx.x * blockDim.x + threadIdx.x;
    if (i >= n) return;
    float zi = wz[i];
    int rank = 0;
    for (int j = 0; j < n; ++j) {
        float zj = wz[j];
        rank += (zj < zi) || (zj == zi && j < i);   // stable argsort rank
    }
    #pragma unroll
    for (int k = 0; k < 8; ++k) gc_out[rank*8 + k] = gc_in[i*8 + k];
    #pragma unroll
    for (int k = 0; k < 4; ++k) ca_out[rank*4 + k] = ca_in[i*4 + k];
}

// ---------------------------------------------------------------------------
// Phase 3: tiled WMMA render. Wave = 16 pixels; per gaussian tile (16):
//   D[16 gaussians][16 pixels] = A(gauss coeffs 16x8) @ B(pixel feats 8x16)
// via two chained V_WMMA_F32_16X16X4_F32. Lane holds one pixel column and 8
// consecutive sorted gaussians -> serial transmittance scan stays in-lane;
// shfl_xor(16) stitches the two halves and carries global T.
// ---------------------------------------------------------------------------
__global__ __launch_bounds__(128) void gs_render(
    const float* __restrict__ gc,    // sorted [n][8]
    const float* __restrict__ ca,    // sorted [n][4]
    const float* __restrict__ bg,
    const float* __restrict__ intrins,
    float* __restrict__ out,         // [128][128][3]
    int n)
{
    const int lane = threadIdx.x & 31;
    const int wave = blockIdx.x * (blockDim.x >> 5) + (threadIdx.x >> 5);
    const int m    = lane & 15;
    const bool hiH = lane >= 16;
    const int pix  = wave * 16 + m;                 // flat pixel = h*128 + w
    const float cx = intrins[2], cy = intrins[3];
    const float X = (float)(pix & 127) + 0.5f - cx;
    const float Y = (float)(pix >> 7)  + 0.5f - cy;

    // B matrix (8x16 pixel features), K-split per lane half:
    // VGPR0: K=0 (f0=X^2) / K=2 (f2=XY); VGPR1: K=1 (f1=Y^2) / K=3 (f3=X)
    v2f bA, bB;
    bA.x = hiH ? X*Y : X*X;
    bA.y = hiH ? X   : Y*Y;
    bB.x = hiH ? 0.f : Y;      // K=4 (f4=Y) / K=6 (pad)
    bB.y = hiH ? 0.f : 1.0f;   // K=5 (f5=1) / K=7 (pad)

    float T = 1.0f;
    float accR = 0.f, accG = 0.f, accB = 0.f;
    const int ntile = n >> 4;

    for (int t = 0; t < ntile; ++t) {
        const int gA = (t << 4) + m;   // A-matrix row (gaussian) for this lane
        const float4 lo4 = reinterpret_cast<const float4*>(gc)[gA*2 + 0];
        const float4 hi4 = reinterpret_cast<const float4*>(gc)[gA*2 + 1];
        v2f aA, aB;
        aA.x = hiH ? lo4.z : lo4.x;    // K=0 (g0) / K=2 (g2)
        aA.y = hiH ? lo4.w : lo4.y;    // K=1 (g1) / K=3 (g3)
        aB.x = hiH ? 0.f  : hi4.x;     // K=4 (g4) / K=6 (0)
        aB.y = hiH ? 0.f  : hi4.y;     // K=5 (g5) / K=7 (0)

        v8f c = {0.f,0.f,0.f,0.f,0.f,0.f,0.f,0.f};
        c = __builtin_amdgcn_wmma_f32_16x16x4_f32(false, aA, false, bA, (short)0, c, false, false);
        c = __builtin_amdgcn_wmma_f32_16x16x4_f32(false, aB, false, bB, (short)0, c, false, false);
        // c[r] = power of gaussian (t*16 + 8*hiH + r) at pixel m

        const int gbase = (t << 4) + (hiH ? 8 : 0);
        float Ploc = 1.0f;
        float lr = 0.f, lg = 0.f, lb = 0.f;
        #pragma unroll
        for (int r = 0; r < 8; ++r) {
            float4 col = reinterpret_cast<const float4*>(ca)[gbase + r];
            float p  = fminf(c[r], 0.0f);
            float gv = __expf(p);
            float ap = fminf(0.99f, col.w * gv);
            float w  = ap * Ploc;
            lr = fmaf(w, col.x, lr);
            lg = fmaf(w, col.y, lg);
            lb = fmaf(w, col.z, lb);
            Ploc *= (1.0f - ap);
        }
        float Poth  = __shfl_xor(Ploc, 16, 32);      // other half's product
        float scale = hiH ? T * Poth : T;            // hi half is later in depth order
        accR = fmaf(scale, lr, accR);
        accG = fmaf(scale, lg, accG);
        accB = fmaf(scale, lb, accB);
        T *= Ploc * Poth;
    }

    accR += __shfl_xor(accR, 16, 32);
    accG += __shfl_xor(accG, 16, 32);
    accB += __shfl_xor(accB, 16, 32);
    if (!hiH) {
        out[pix*3 + 0] = fmaf(T, bg[0], accR);
        out[pix*3 + 1] = fmaf(T, bg[1], accG);
        out[pix*3 + 2] = fmaf(T, bg[2], accB);
    }
}

extern "C" void kernel_launch(void* const* d_in, const int* in_sizes, int n_in,
                              void* d_out, int out_size, void* d_ws, size_t ws_size,
                              hipStream_t stream)
{
    const float* mean      = (const float*)d_in[0];
    const float* qvec      = (const float*)d_in[1];
    const float* svec_raw  = (const float*)d_in[2];
    const float* color_raw = (const float*)d_in[3];
    const float* alpha_raw = (const float*)d_in[4];
    const float* bg        = (const float*)d_in[5];
    const float* viewmat   = (const float*)d_in[6];
    const float* intrins   = (const float*)d_in[7];
    float* out = (float*)d_out;

    const int n = in_sizes[0] / 3;        // 2048 gaussians
    float* ws   = (float*)d_ws;
    float* wz   = ws;                     // [n]
    float* wgcu = ws + n;                 // [n][8]
    float* wcau = ws + 9*n;               // [n][4]
    float* wgcs = ws + 13*n;              // [n][8] sorted
    float* wcas = ws + 21*n;              // [n][4] sorted
    (void)n_in; (void)ws_size;

    dim3 b1(256), g1((n + 255) / 256);
    hipLaunchKernelGGL(gs_prep, g1, b1, 0, stream,
                       mean, qvec, svec_raw, color_raw, alpha_raw, viewmat, intrins,
                       wz, wgcu, wcau, n);
    hipLaunchKernelGGL(gs_sort, g1, b1, 0, stream, wz, wgcu, wcau, wgcs, wcas, n);

    const int npix  = out_size / 3;       // 16384 pixels
    const int waves = npix / 16;          // 1024 waves of 16 pixels
    dim3 b3(128), g3(waves / 4);          // 4 waves per block
    hipLaunchKernelGGL(gs_render, g3, b3, 0, stream, wgcs, wcas, bg, intrins, out, n);
}